// QuantMapReduce_9964324127047
// MI455X (gfx1250) — compile-verified
//
#include <hip/hip_runtime.h>

// out[b,hw] = sum_c wq[c] * x[b,c,hw] + bias     (1x1 conv, C=128 -> 1)
//
// Bandwidth-bound: 1.07 GB read @ 23.3 TB/s ~= 46 us floor, 0.54 GFLOP is free.
// Reduction rides the matrix pipe with V_WMMA_F32_16X16X4_F32 (full f32
// precision): A = 16 pixels x 4 channels of x, B = wq[c..c+3] broadcast to all
// 16 columns, so every column of the 16x16 f32 accumulator equals the per-pixel
// dot product. 32 WMMA steps cover K=128; 4 independent accumulators per wave
// (64 pixels) hide WMMA->WMMA RAW latency.
//
// Prefetch is UNGUARDED: GLOBAL_PREFETCH_B8 is speculative (ISA 10.5) -- a
// failed translation is silently dropped, so overshooting the tensor tail is
// safe and we avoid per-iteration SALU compare/branch chains in the hot loop.

typedef float v2f __attribute__((ext_vector_type(2)));
typedef float v4f __attribute__((ext_vector_type(4)));
typedef float v8f __attribute__((ext_vector_type(8)));

#define C_CH   128
#define HW_TOT (512 * 512)          // per-batch spatial size (channel stride)
#define TILES  4                    // 16-pixel WMMA tiles per wave
#define PIX_PER_WAVE (16 * TILES)   // 64 pixels per wave
#define BLOCK_THREADS 256           // 8 waves (wave32) per block
#define PREFETCH_CH 16              // prefetch lead, in channels

__global__ __launch_bounds__(BLOCK_THREADS)
void conv1x1_qdq_wmma(const float* __restrict__ x,
                      const float* __restrict__ W,
                      const float* __restrict__ bvec,
                      float* __restrict__ out,
                      int npix)     // total output pixels = B*H*W
{
    // ---- on-device quant-dequant of the 128 weights into LDS -------------
    __shared__ float wq[C_CH];
    const int tid = threadIdx.x;
    if (tid < C_CH) {
        float m = 0.0f;
        #pragma unroll 8
        for (int c = 0; c < C_CH; ++c) m = fmaxf(m, fabsf(W[c]));
        const float scale = m * (1.0f / 127.0f);
        const float q = fminf(fmaxf(rintf(W[tid] / scale), -127.0f), 127.0f);
        wq[tid] = q * scale;
    }
    __syncthreads();

    // ---- wave/lane geometry ---------------------------------------------
    const int lane = tid & 31;
    const int half = lane >> 4;     // 0: lanes 0-15 (K0/K1), 1: lanes 16-31 (K2/K3)
    const int pl   = lane & 15;     // pixel (M row) within a 16-pixel tile
    const int wave = tid >> 5;
    const int group = blockIdx.x * (BLOCK_THREADS >> 5) + wave;
    const long long p0 = (long long)group * PIX_PER_WAVE;
    if (p0 >= npix) return;         // wave-uniform: EXEC stays all-ones for WMMA

    const int b  = (int)(p0 / HW_TOT);
    const int hw = (int)(p0 % HW_TOT);
    const float* xb = x + (size_t)b * C_CH * HW_TOT + hw;

    v8f acc[TILES];
    #pragma unroll
    for (int t = 0; t < TILES; ++t) acc[t] = {};

    // ---- K loop: 32 chunks of 4 channels --------------------------------
    // ISA layout (16x4 f32 A): lanes0-15 VGPR0=K0,VGPR1=K1; lanes16-31 K2,K3.
    // This lane's A components are channels (c0, c0+1) with c0 = 4i + 2*half;
    // B components are wq at the same channels (all 16 columns identical).
    #pragma unroll 4
    for (int i = 0; i < C_CH / 4; ++i) {
        const int c0 = 4 * i + 2 * half;
        v2f bm;
        bm.x = wq[c0];
        bm.y = wq[c0 + 1];
        const float* xc = xb + (size_t)c0 * HW_TOT;

        // speculative L2 prime, no guard needed (dropped if untranslatable)
        __builtin_prefetch(xc + (size_t)PREFETCH_CH * HW_TOT + pl, 0, 1);

        #pragma unroll
        for (int t = 0; t < TILES; ++t) {
            v2f am;
            am.x = xc[pl + 16 * t];                  // channel c0,   pixel pl
            am.y = xc[(size_t)HW_TOT + pl + 16 * t]; // channel c0+1, pixel pl
            acc[t] = __builtin_amdgcn_wmma_f32_16x16x4_f32(
                false, am, false, bm, (short)0, acc[t], false, false);
        }
    }

    // ---- store: all 16 columns of D are identical ------------------------
    // lane 0 holds rows 0-7 of col 0, lane 16 holds rows 8-15 of col 0.
    const float bias = bvec[0];
    if (pl == 0) {
        #pragma unroll
        for (int t = 0; t < TILES; ++t) {
            float* op = out + p0 + 16 * t + half * 8;
            v4f lo = { acc[t][0] + bias, acc[t][1] + bias,
                       acc[t][2] + bias, acc[t][3] + bias };
            v4f hi = { acc[t][4] + bias, acc[t][5] + bias,
                       acc[t][6] + bias, acc[t][7] + bias };
            *(v4f*)(op)     = lo;
            *(v4f*)(op + 4) = hi;
        }
    }
}

extern "C" void kernel_launch(void* const* d_in, const int* in_sizes, int n_in,
                              void* d_out, int out_size, void* d_ws, size_t ws_size,
                              hipStream_t stream) {
    const float* x = (const float*)d_in[0];   // (8,128,512,512) f32
    const float* W = (const float*)d_in[1];   // (1,128,1,1)     f32
    const float* b = (const float*)d_in[2];   // (1,)            f32
    float* out = (float*)d_out;               // (8,1,512,512)   f32

    const int npix = out_size;                // 8*512*512 = 2,097,152
    const int waves = (npix + PIX_PER_WAVE - 1) / PIX_PER_WAVE;   // 32768
    const int blocks = (waves + (BLOCK_THREADS / 32) - 1) / (BLOCK_THREADS / 32); // 4096

    conv1x1_qdq_wmma<<<blocks, BLOCK_THREADS, 0, stream>>>(x, W, b, out, npix);
}